// PathFinder_2336462209714
// MI455X (gfx1250) — compile-verified
//
#include <hip/hip_runtime.h>
#include <hip/hip_bf16.h>

typedef __attribute__((ext_vector_type(2))) float v2f;
typedef __attribute__((ext_vector_type(8))) float v8f;

#define HID   4096
#define BATCH 8
#define NNODE 200
#define NEDGE 400
#define TOPK  3

// ---------------------------------------------------------------------------
// Zero a buffer of n floats (used for the padded-A zero row).
// ---------------------------------------------------------------------------
__global__ __launch_bounds__(256)
void pf_zero(float* __restrict__ p, int n)
{
    const int i = blockIdx.x * 256 + threadIdx.x;
    if (i < n) p[i] = 0.0f;
}

// ---------------------------------------------------------------------------
// out[M, HID] = X[M, HID] @ W[HID, HID]^T + bias   (optionally ReLU)
// One wave per 16x16 output tile, f32 WMMA 16x16x4 over K = HID.
// Rows past M read from `zrow` (4096 zeros) so EXEC stays all-ones and no
// per-iteration masking math is needed.
// blockDim = 256 (8 waves); grid = (HID/128, ceil(M/16)).
// ---------------------------------------------------------------------------
__global__ __launch_bounds__(256)
void pf_gemm_xwT(const float* __restrict__ X, const float* __restrict__ W,
                 const float* __restrict__ bias, const float* __restrict__ zrow,
                 float* __restrict__ out, int M, int relu)
{
    const int lane = threadIdx.x & 31;
    const int wave = threadIdx.x >> 5;
    const int half = lane >> 4;        // 0: lanes 0-15, 1: lanes 16-31
    const int l16  = lane & 15;

    const int n0 = (blockIdx.x * 8 + wave) * 16;   // output column tile
    const int m0 = blockIdx.y * 16;                // output row tile

    // A fragment: lane -> row m0+l16, VGPR pair -> K = k + 2*half + {0,1}
    const int rowm = m0 + l16;
    const float* __restrict__ xrow =
        (rowm < M) ? (X + (size_t)rowm * HID) : zrow;  // zero-padded rows
    // B fragment (W^T): lane -> col n0+l16, VGPR pair -> K = k + 2*half + {0,1}
    const float* __restrict__ wrow = W + (size_t)(n0 + l16) * HID;

    v8f c = {};
    for (int k = 0; k < HID; k += 16) {
        // Speculative prefetch (TH=0): safe past end of buffer, no guard.
        __builtin_prefetch(wrow + k + 1024, 0, 0);     // global_prefetch_b8
        #pragma unroll
        for (int kk = 0; kk < 16; kk += 4) {
            const int ko = k + kk + 2 * half;
            const v2f a = *(const v2f*)(xrow + ko);
            const v2f b = *(const v2f*)(wrow + ko);
            // D = A x B + C, fp32 16x16x4 (EXEC all-ones: no divergence here)
            c = __builtin_amdgcn_wmma_f32_16x16x4_f32(
                    false, a, false, b, (short)0, c, false, false);
        }
    }

    const float bn = bias[n0 + l16];
    #pragma unroll
    for (int i = 0; i < 8; ++i) {
        // C/D layout: VGPR i -> M = i (lanes 0-15) or 8+i (lanes 16-31), N = l16
        const int m = m0 + i + half * 8;
        if (m < M) {
            float v = c[i] + bn;
            if (relu) v = fmaxf(v, 0.0f);
            out[(size_t)m * HID + (n0 + l16)] = v;
        }
    }
}

// ---------------------------------------------------------------------------
// L2 norm of each row of a [rows, HID] matrix.
// ---------------------------------------------------------------------------
__global__ __launch_bounds__(256)
void pf_rownorm(const float* __restrict__ X, float* __restrict__ nrm)
{
    const int r = blockIdx.x;
    const float* p = X + (size_t)r * HID;
    float s = 0.0f;
    for (int i = threadIdx.x; i < HID; i += 256) { float v = p[i]; s += v * v; }
    __shared__ float red[256];
    red[threadIdx.x] = s; __syncthreads();
    for (int st = 128; st > 0; st >>= 1) {
        if (threadIdx.x < st) red[threadIdx.x] += red[threadIdx.x + st];
        __syncthreads();
    }
    if (threadIdx.x == 0) nrm[r] = sqrtf(red[0]);
}

// ---------------------------------------------------------------------------
// Masked cosine similarity sim[b,n]; grid = (NNODE, BATCH).
// ---------------------------------------------------------------------------
__global__ __launch_bounds__(256)
void pf_sim(const float* __restrict__ qp, const float* __restrict__ npj,
            const float* __restrict__ qn, const float* __restrict__ nn,
            const int* __restrict__ batch, float* __restrict__ sim)
{
    const int n = blockIdx.x, b = blockIdx.y;
    const float* q = qp  + (size_t)b * HID;
    const float* p = npj + (size_t)n * HID;
    float s = 0.0f;
    for (int i = threadIdx.x; i < HID; i += 256) s += q[i] * p[i];
    __shared__ float red[256];
    red[threadIdx.x] = s; __syncthreads();
    for (int st = 128; st > 0; st >>= 1) {
        if (threadIdx.x < st) red[threadIdx.x] += red[threadIdx.x + st];
        __syncthreads();
    }
    if (threadIdx.x == 0) {
        const float denom = fmaxf(qn[b] * nn[n], 1e-8f);
        const float v = red[0] / denom;
        sim[b * NNODE + n] = (batch[n] == b) ? v : -__builtin_inff();
    }
}

// ---------------------------------------------------------------------------
// Per-batch: stable top-3 starts, first valid outgoing edge, counts.
// One wave; thread b < BATCH does batch b serially (tiny work).
// ---------------------------------------------------------------------------
__global__ __launch_bounds__(32)
void pf_select(const float* __restrict__ sim, const int* __restrict__ edge_index,
               const int* __restrict__ batch,
               int* __restrict__ starts, int* __restrict__ dsel,
               int* __restrict__ has, int* __restrict__ cnt,
               int* __restrict__ cntn)
{
    const int b = threadIdx.x;
    if (b >= BATCH) return;
    const float* s = sim + b * NNODE;

    int chosen[TOPK];
    for (int t = 0; t < TOPK; ++t) {
        float best = 0.0f; int bi = 0; bool found = false;
        for (int n = 0; n < NNODE; ++n) {
            bool used = false;
            for (int u = 0; u < t; ++u) used |= (chosen[u] == n);
            if (used) continue;
            const float v = s[n];
            if (!found || v > best) { best = v; bi = n; found = true; } // ties -> lowest index (jax top_k)
        }
        chosen[t] = bi;
    }

    const int* src = edge_index;
    const int* dst = edge_index + NEDGE;
    int c = 0;
    for (int t = 0; t < TOPK; ++t) {
        const int st = chosen[t];
        int first = NEDGE;
        for (int e = 0; e < NEDGE; ++e) {
            if (src[e] == st && batch[dst[e]] == b) { first = e; break; }
        }
        const int h = (first < NEDGE) ? 1 : 0;
        starts[b * TOPK + t] = st;
        dsel  [b * TOPK + t] = dst[min(first, NEDGE - 1)];
        has   [b * TOPK + t] = h;
        c += h;
    }
    cnt[b] = c;
    int cn = 0;
    for (int n = 0; n < NNODE; ++n) cn += (batch[n] == b) ? 1 : 0;
    cntn[b] = cn;
}

// ---------------------------------------------------------------------------
// agg[b,h] = mean over valid paths of 0.5*(nf[start]+nf[dst])
// fallback[b,h] = mean of npj over nodes in batch b.  grid=(HID/256, BATCH).
// ---------------------------------------------------------------------------
__global__ __launch_bounds__(256)
void pf_agg_fb(const float* __restrict__ nf, const float* __restrict__ npj,
               const int* __restrict__ batch,
               const int* __restrict__ starts, const int* __restrict__ dsel,
               const int* __restrict__ has, const int* __restrict__ cnt,
               const int* __restrict__ cntn,
               float* __restrict__ agg, float* __restrict__ fallback)
{
    const int h = blockIdx.x * 256 + threadIdx.x;
    const int b = blockIdx.y;

    float s = 0.0f;
    #pragma unroll
    for (int t = 0; t < TOPK; ++t) {
        if (has[b * TOPK + t]) {
            s += 0.5f * (nf[(size_t)starts[b * TOPK + t] * HID + h] +
                         nf[(size_t)dsel  [b * TOPK + t] * HID + h]);
        }
    }
    agg[(size_t)b * HID + h] = s / (float)max(cnt[b], 1);

    float f = 0.0f;
    for (int n = 0; n < NNODE; ++n)
        if (batch[n] == b) f += npj[(size_t)n * HID + h];
    fallback[(size_t)b * HID + h] = f / (float)max(cntn[b], 1);
}

// ---------------------------------------------------------------------------
// pr = cnt>0 ? h2 : fallback; then LayerNorm(eps=1e-5)*g + b.  grid = BATCH.
// ---------------------------------------------------------------------------
__global__ __launch_bounds__(256)
void pf_finalize(const float* __restrict__ h2, const float* __restrict__ fallback,
                 const int* __restrict__ cnt,
                 const float* __restrict__ g, const float* __restrict__ be,
                 float* __restrict__ out)
{
    const int b = blockIdx.x;
    const float* src = (cnt[b] > 0) ? (h2 + (size_t)b * HID)
                                    : (fallback + (size_t)b * HID);
    float v[HID / 256];
    float s = 0.0f;
    #pragma unroll
    for (int i = 0; i < HID / 256; ++i) {
        v[i] = src[i * 256 + threadIdx.x];
        s += v[i];
    }
    __shared__ float red[256];
    __shared__ float stat[2];
    red[threadIdx.x] = s; __syncthreads();
    for (int st = 128; st > 0; st >>= 1) {
        if (threadIdx.x < st) red[threadIdx.x] += red[threadIdx.x + st];
        __syncthreads();
    }
    if (threadIdx.x == 0) stat[0] = red[0] / (float)HID;
    __syncthreads();
    const float mu = stat[0];

    float s2 = 0.0f;
    #pragma unroll
    for (int i = 0; i < HID / 256; ++i) { const float d = v[i] - mu; s2 += d * d; }
    red[threadIdx.x] = s2; __syncthreads();
    for (int st = 128; st > 0; st >>= 1) {
        if (threadIdx.x < st) red[threadIdx.x] += red[threadIdx.x + st];
        __syncthreads();
    }
    if (threadIdx.x == 0) stat[1] = rsqrtf(red[0] / (float)HID + 1e-5f);
    __syncthreads();
    const float rstd = stat[1];

    #pragma unroll
    for (int i = 0; i < HID / 256; ++i) {
        const int h = i * 256 + threadIdx.x;
        out[(size_t)b * HID + h] = (v[i] - mu) * rstd * g[h] + be[h];
    }
}

// ---------------------------------------------------------------------------
extern "C" void kernel_launch(void* const* d_in, const int* in_sizes, int n_in,
                              void* d_out, int out_size, void* d_ws, size_t ws_size,
                              hipStream_t stream)
{
    (void)in_sizes; (void)n_in; (void)out_size; (void)ws_size;
    const float* query = (const float*)d_in[0];
    const float* nf    = (const float*)d_in[1];
    const float* Wq    = (const float*)d_in[2];
    const float* bq    = (const float*)d_in[3];
    const float* Wn    = (const float*)d_in[4];
    const float* bn    = (const float*)d_in[5];
    const float* W1    = (const float*)d_in[6];
    const float* b1    = (const float*)d_in[7];
    const float* W2    = (const float*)d_in[8];
    const float* b2    = (const float*)d_in[9];
    const float* ln_g  = (const float*)d_in[10];
    const float* ln_b  = (const float*)d_in[11];
    const int*   eidx  = (const int*)d_in[12];
    const int*   batch = (const int*)d_in[13];
    float* out = (float*)d_out;

    // workspace layout (fp32 elements)
    float* ws = (float*)d_ws;
    float* qp   = ws;                         // 8*4096
    float* npj  = qp   + BATCH * HID;         // 200*4096
    float* sim  = npj  + NNODE * HID;         // 8*200
    float* qn   = sim  + BATCH * NNODE;       // 8
    float* nn   = qn   + BATCH;               // 200
    float* agg  = nn   + NNODE;               // 8*4096
    float* hmid = agg  + BATCH * HID;         // 8*4096
    float* h2   = hmid + BATCH * HID;         // 8*4096
    float* fb   = h2   + BATCH * HID;         // 8*4096
    float* zrow = fb   + BATCH * HID;         // 4096 zeros (padded-A row)
    int*   iws    = (int*)(zrow + HID);
    int*   starts = iws;                      // 8*3
    int*   dsel   = starts + BATCH * TOPK;    // 8*3
    int*   has    = dsel   + BATCH * TOPK;    // 8*3
    int*   cnt    = has    + BATCH * TOPK;    // 8
    int*   cntn   = cnt    + BATCH;           // 8

    const dim3 blk256(256), blkWave(32);
    const dim3 gGemmQ(HID / 128, 1);                    // M = 8  (1 row tile)
    const dim3 gGemmN(HID / 128, (NNODE + 15) / 16);    // M = 200 (13 row tiles)

    // 0: zero row for padded A-fragment rows
    pf_zero<<<dim3(HID / 256), blk256, 0, stream>>>(zrow, HID);

    // 1-2: projections (WMMA f32 streaming GEMMs, HBM-bound on weights)
    pf_gemm_xwT<<<gGemmQ, blk256, 0, stream>>>(query, Wq, bq, zrow, qp, BATCH, 0);
    pf_gemm_xwT<<<gGemmN, blk256, 0, stream>>>(nf,    Wn, bn, zrow, npj, NNODE, 0);

    // 3: norms + masked cosine similarity
    pf_rownorm<<<dim3(BATCH), blk256, 0, stream>>>(qp,  qn);
    pf_rownorm<<<dim3(NNODE), blk256, 0, stream>>>(npj, nn);
    pf_sim<<<dim3(NNODE, BATCH), blk256, 0, stream>>>(qp, npj, qn, nn, batch, sim);

    // 4: top-3 starts + first valid edge per start
    pf_select<<<dim3(1), blkWave, 0, stream>>>(sim, eidx, batch,
                                               starts, dsel, has, cnt, cntn);

    // 5: path aggregation + per-batch fallback mean
    pf_agg_fb<<<dim3(HID / 256, BATCH), blk256, 0, stream>>>(
        nf, npj, batch, starts, dsel, has, cnt, cntn, agg, fb);

    // 6-7: path-aggregator MLP (WMMA GEMMs)
    pf_gemm_xwT<<<gGemmQ, blk256, 0, stream>>>(agg,  W1, b1, zrow, hmid, BATCH, 1);
    pf_gemm_xwT<<<gGemmQ, blk256, 0, stream>>>(hmid, W2, b2, zrow, h2,   BATCH, 0);

    // 8: select h vs fallback, LayerNorm
    pf_finalize<<<dim3(BATCH), blk256, 0, stream>>>(h2, fb, cnt, ln_g, ln_b, out);
}